// GatedGAT_62225486185197
// MI455X (gfx1250) — compile-verified
//
#include <hip/hip_runtime.h>
#include <math.h>

typedef float v2f __attribute__((ext_vector_type(2)));
typedef float v8f __attribute__((ext_vector_type(8)));

#define NNODES 50000
#define NEDGES 800000
#define INFEAT 128
#define OUTF   64
#define MAPF   64
#define NH     4

// ---------------------------------------------------------------------------
// float atomic max via signed/unsigned int ordering trick (init to -inf).
// ---------------------------------------------------------------------------
__device__ __forceinline__ void atomicMaxF(float* addr, float val) {
    if (val >= 0.0f) atomicMax((int*)addr, __float_as_int(val));
    else             atomicMin((unsigned int*)addr, __float_as_uint(val));
}

// ---------------------------------------------------------------------------
// fill kernel
// ---------------------------------------------------------------------------
__global__ void k_fill(float* __restrict__ p, float v, long n) {
    long i = (long)blockIdx.x * blockDim.x + threadIdx.x;
    if (i < n) p[i] = v;
}

// ---------------------------------------------------------------------------
// WMMA fp32 GEMM:  C[M,Ncols] = concat(A0[M,K0], A1[M,K1]) @ W[K,Ncols] + bias
// Block = 128 threads = 4 wave32s. Each wave computes one 16x16 tile.
// blockIdx.x -> 16-row tile, blockIdx.y*4 + wave -> 16-col tile.
// A-panel (16 x K, K<=192) staged into LDS with CDNA5 async global->LDS
// B128 copies (ASYNCcnt), then consumed by V_WMMA_F32_16X16X4_F32.
// ---------------------------------------------------------------------------
__global__ __launch_bounds__(128)
void k_wmma_gemm(const float* __restrict__ A0, int K0,
                 const float* __restrict__ A1, int K1,
                 const float* __restrict__ W, const float* __restrict__ bias,
                 float* __restrict__ C, int M, int Ncols) {
    __shared__ float lds[16 * 192];
    const int K = K0 + K1;
    const int m0 = blockIdx.x * 16;

    // ---- async stage of the 16xK A panel: 4 floats (16B) per lane per chunk.
    // K is a multiple of 4 and the concat boundary (K0=128) is 4-aligned, so
    // every 16B chunk lies in one row of one source and is 16B aligned.
    const int totalChunks = (16 * K) >> 2;          // 512 (K=128) / 768 (K=192)
    for (int ch = threadIdx.x; ch < totalChunks; ch += blockDim.x) {
        int li = ch << 2;                            // linear float index
        int r = li / K, c = li - r * K;
        int row = m0 + r;
        if (row >= M) row = M - 1;                   // defensive (never hit: M%16==0)
        const float* gptr = (c < K0) ? (A0 + (long)row * K0 + c)
                                     : (A1 + (long)row * K1 + (c - K0));
        unsigned lds_addr = (unsigned)(unsigned long long)(const void*)(lds + (r * K + c));
        unsigned long long ga = (unsigned long long)gptr;
        // GLOBAL_LOAD_ASYNC_TO_LDS_B128: copies 16B from each lane's global address to its LDS address
        asm volatile("global_load_async_to_lds_b128 %0, %1, off"
                     :: "v"(lds_addr), "v"(ga) : "memory");
    }
    asm volatile("s_wait_asynccnt 0x0" ::: "memory");
    __syncthreads();

    const int wave = threadIdx.x >> 5;
    const int lane = threadIdx.x & 31;
    const int half = lane >> 4;      // 0: lanes 0-15, 1: lanes 16-31
    const int lq   = lane & 15;
    const int col0 = (blockIdx.y * 4 + wave) * 16;
    if (col0 >= Ncols) return;       // wave-uniform exit (EXEC stays full per wave)

    v8f acc = {};
    for (int k0 = 0; k0 < K; k0 += 4) {
        const int kk = k0 + 2 * half;          // ISA 16x4 f32 A layout
        v2f a, b;
        a.x = lds[lq * K + kk];
        a.y = lds[lq * K + kk + 1];
        b.x = W[(long)kk * Ncols + col0 + lq];       // 4x16 f32 B layout
        b.y = W[(long)(kk + 1) * Ncols + col0 + lq];
        acc = __builtin_amdgcn_wmma_f32_16x16x4_f32(
            /*neg_a=*/false, a, /*neg_b=*/false, b,
            /*c_mod=*/(short)0, acc, /*reuse_a=*/false, /*reuse_b=*/false);
    }

    // C/D layout: VGPR r -> row m0 + 8*half + r, col = col0 + lq
    const int col = col0 + lq;
    const float bv = bias ? bias[col] : 0.0f;
#pragma unroll
    for (int r = 0; r < 8; ++r) {
        int row = m0 + half * 8 + r;
        if (row < M) C[(long)row * Ncols + col] = acc[r] + bv;
    }
}

// ---------------------------------------------------------------------------
// el/er:  el[n,h] = dot(feat[n,h,:], attn_l[h,:])   (and er with attn_r)
// ---------------------------------------------------------------------------
__global__ void k_el_er(const float* __restrict__ feat,
                        const float* __restrict__ attn_l,
                        const float* __restrict__ attn_r,
                        float* __restrict__ el, float* __restrict__ er) {
    int i = blockIdx.x * blockDim.x + threadIdx.x;
    if (i >= NNODES * NH) return;
    int n = i >> 2, h = i & 3;
    const float* f = feat + (long)n * (NH * OUTF) + h * OUTF;
    const float* al = attn_l + h * OUTF;
    const float* ar = attn_r + h * OUTF;
    float sl = 0.f, sr = 0.f;
    for (int d = 0; d < OUTF; ++d) { float v = f[d]; sl += v * al[d]; sr += v * ar[d]; }
    el[i] = sl; er[i] = sr;
}

// ---------------------------------------------------------------------------
// edge pass A: degree, leaky-relu attention logits, per-dst running max
// ---------------------------------------------------------------------------
__global__ void k_edgeA(const int* __restrict__ src, const int* __restrict__ dst,
                        const float* __restrict__ el, const float* __restrict__ er,
                        float* __restrict__ deg, float* __restrict__ ebuf,
                        float* __restrict__ emax) {
    int e = blockIdx.x * blockDim.x + threadIdx.x;
    if (e >= NEDGES) return;
    int s = src[e], d = dst[e];
    atomicAdd(&deg[d], 1.0f);
#pragma unroll
    for (int h = 0; h < NH; ++h) {
        float t = el[s * NH + h] + er[d * NH + h];
        t = (t > 0.0f) ? t : 0.2f * t;           // leaky_relu(0.2)
        ebuf[e * NH + h] = t;
        atomicMaxF(&emax[d * NH + h], t);
    }
}

// ---------------------------------------------------------------------------
// edge pass B: sum of x[src] into sumx[dst] (c<128), max of z[src] into maxz
// one thread per (edge, c) with c in [0,192)
// ---------------------------------------------------------------------------
__global__ void k_edgeB(const int* __restrict__ src, const int* __restrict__ dst,
                        const float* __restrict__ x, const float* __restrict__ z,
                        float* __restrict__ sumx, float* __restrict__ maxz) {
    long i = (long)blockIdx.x * blockDim.x + threadIdx.x;
    if (i >= (long)NEDGES * 192) return;
    int e = (int)(i / 192);
    int c = (int)(i - (long)e * 192);
    int s = src[e], d = dst[e];
    if (c < INFEAT) {
        atomicAdd(&sumx[(long)d * INFEAT + c], x[(long)s * INFEAT + c]);
    } else {
        int f = c - INFEAT;
        atomicMaxF(&maxz[(long)d * MAPF + f], z[(long)s * MAPF + f]);
    }
}

// ---------------------------------------------------------------------------
// edge pass C: ew = exp(e - emax[dst]); accumulate denominator
// ---------------------------------------------------------------------------
__global__ void k_edgeC(const int* __restrict__ dst, float* __restrict__ ebuf,
                        const float* __restrict__ emax, float* __restrict__ esum) {
    int i = blockIdx.x * blockDim.x + threadIdx.x;
    if (i >= NEDGES * NH) return;
    int e = i >> 2, h = i & 3;
    int d = dst[e];
    float w = __expf(ebuf[i] - emax[d * NH + h]);
    ebuf[i] = w;
    atomicAdd(&esum[d * NH + h], w);
}

// ---------------------------------------------------------------------------
// edge pass D: alpha = ew / esum[dst]
// ---------------------------------------------------------------------------
__global__ void k_edgeD(const int* __restrict__ dst, float* __restrict__ ebuf,
                        const float* __restrict__ esum) {
    int i = blockIdx.x * blockDim.x + threadIdx.x;
    if (i >= NEDGES * NH) return;
    int e = i >> 2, h = i & 3;
    ebuf[i] = ebuf[i] / esum[dst[e] * NH + h];
}

// ---------------------------------------------------------------------------
// edge pass E: attnsum[dst, h, d] += alpha[e,h] * feat[src, h, d]
// one thread per (edge, c) with c in [0,256)
// ---------------------------------------------------------------------------
__global__ void k_edgeE(const int* __restrict__ src, const int* __restrict__ dst,
                        const float* __restrict__ ebuf, const float* __restrict__ feat,
                        float* __restrict__ attnsum) {
    long i = (long)blockIdx.x * blockDim.x + threadIdx.x;
    if (i >= (long)NEDGES * 256) return;
    int e = (int)(i >> 8);
    int c = (int)(i & 255);
    int h = c >> 6;
    int s = src[e], d = dst[e];
    atomicAdd(&attnsum[(long)d * 256 + c], ebuf[e * NH + h] * feat[(long)s * 256 + c]);
}

// ---------------------------------------------------------------------------
// node epilogue: mean_x, maxz fixup, gate = sigmoid([x,maxz,meanx]@W_gate+b),
// gated = mean_h( gate[h] * (attnsum[h,:] + b_gat[h,:]) )
// ---------------------------------------------------------------------------
__global__ void k_node(const float* __restrict__ x, const float* __restrict__ sumx,
                       const float* __restrict__ maxz, const float* __restrict__ deg,
                       const float* __restrict__ W_gate, const float* __restrict__ b_gate,
                       const float* __restrict__ attnsum, const float* __restrict__ b_gat,
                       float* __restrict__ gated) {
    int n = blockIdx.x * blockDim.x + threadIdx.x;
    if (n >= NNODES) return;
    float dg = deg[n];
    float inv = 1.0f / fmaxf(dg, 1.0f);
    bool hasdeg = dg > 0.0f;

    float g[NH];
#pragma unroll
    for (int h = 0; h < NH; ++h) g[h] = b_gate[h];

    const float* xr = x + (long)n * INFEAT;
    for (int i = 0; i < INFEAT; ++i) {
        float v = xr[i];
#pragma unroll
        for (int h = 0; h < NH; ++h) g[h] += v * W_gate[i * NH + h];
    }
    for (int i = 0; i < MAPF; ++i) {
        float v = hasdeg ? maxz[(long)n * MAPF + i] : 0.0f;
#pragma unroll
        for (int h = 0; h < NH; ++h) g[h] += v * W_gate[(INFEAT + i) * NH + h];
    }
    for (int i = 0; i < INFEAT; ++i) {
        float v = sumx[(long)n * INFEAT + i] * inv;
#pragma unroll
        for (int h = 0; h < NH; ++h) g[h] += v * W_gate[(INFEAT + MAPF + i) * NH + h];
    }
#pragma unroll
    for (int h = 0; h < NH; ++h) g[h] = 1.0f / (1.0f + __expf(-g[h]));

    for (int d = 0; d < OUTF; ++d) {
        float acc = 0.0f;
#pragma unroll
        for (int h = 0; h < NH; ++h) {
            float ao = attnsum[(long)n * 256 + h * OUTF + d] + b_gat[h * OUTF + d];
            acc += g[h] * ao;
        }
        gated[(long)n * OUTF + d] = acc * 0.25f;   // mean over 4 heads
    }
}

// ---------------------------------------------------------------------------
// host-side launch
// ---------------------------------------------------------------------------
extern "C" void kernel_launch(void* const* d_in, const int* in_sizes, int n_in,
                              void* d_out, int out_size, void* d_ws, size_t ws_size,
                              hipStream_t stream) {
    const float* x       = (const float*)d_in[0];
    const int*   eidx    = (const int*)d_in[1];
    const float* W_gm    = (const float*)d_in[2];
    const float* b_gm    = (const float*)d_in[3];
    const float* W_gate  = (const float*)d_in[4];
    const float* b_gate  = (const float*)d_in[5];
    const float* W_fc    = (const float*)d_in[6];
    const float* attn_l  = (const float*)d_in[7];
    const float* attn_r  = (const float*)d_in[8];
    const float* b_gat   = (const float*)d_in[9];
    const float* W_merge = (const float*)d_in[10];
    const float* b_merge = (const float*)d_in[11];
    const int* src = eidx;
    const int* dst = eidx + NEDGES;
    float* out = (float*)d_out;

    // workspace layout (floats)
    float* ws = (float*)d_ws;
    long off = 0;
    float* feat    = ws + off; off += (long)NNODES * 256;   // x @ W_fc
    float* z       = ws + off; off += (long)NNODES * MAPF;  // x @ W_gm + b
    float* el      = ws + off; off += (long)NNODES * NH;
    float* er      = ws + off; off += (long)NNODES * NH;
    float* ebuf    = ws + off; off += (long)NEDGES * NH;    // logits -> ew -> alpha
    float* gated   = ws + off; off += (long)NNODES * OUTF;
    // zero-init block (contiguous)
    float* deg     = ws + off; off += NNODES;
    float* sumx    = ws + off; off += (long)NNODES * INFEAT;
    float* esum    = ws + off; off += (long)NNODES * NH;
    float* attnsum = ws + off; off += (long)NNODES * 256;
    long zero_n    = NNODES + (long)NNODES * INFEAT + (long)NNODES * NH + (long)NNODES * 256;
    // -inf init block (contiguous)
    float* maxz    = ws + off; off += (long)NNODES * MAPF;
    float* emax    = ws + off; off += (long)NNODES * NH;
    long ninf_n    = (long)NNODES * MAPF + (long)NNODES * NH;

    const int TB = 256;
    // init accumulators
    k_fill<<<(int)((zero_n + TB - 1) / TB), TB, 0, stream>>>(deg, 0.0f, zero_n);
    k_fill<<<(int)((ninf_n + TB - 1) / TB), TB, 0, stream>>>(maxz, -INFINITY, ninf_n);

    // dense GEMMs on WMMA path (N = 50000 = 3125 * 16 exactly)
    dim3 gemmBlk(128, 1, 1);
    k_wmma_gemm<<<dim3(3125, 4), gemmBlk, 0, stream>>>(
        x, INFEAT, nullptr, 0, W_fc, nullptr, feat, NNODES, NH * OUTF);   // feat
    k_wmma_gemm<<<dim3(3125, 1), gemmBlk, 0, stream>>>(
        x, INFEAT, nullptr, 0, W_gm, b_gm, z, NNODES, MAPF);              // z

    // attention logit projections
    k_el_er<<<(NNODES * NH + TB - 1) / TB, TB, 0, stream>>>(feat, attn_l, attn_r, el, er);

    // edge phase
    k_edgeA<<<(NEDGES + TB - 1) / TB, TB, 0, stream>>>(src, dst, el, er, deg, ebuf, emax);
    long nB = (long)NEDGES * 192;
    k_edgeB<<<(int)((nB + TB - 1) / TB), TB, 0, stream>>>(src, dst, x, z, sumx, maxz);
    k_edgeC<<<(NEDGES * NH + TB - 1) / TB, TB, 0, stream>>>(dst, ebuf, emax, esum);
    k_edgeD<<<(NEDGES * NH + TB - 1) / TB, TB, 0, stream>>>(dst, ebuf, esum);
    long nE = (long)NEDGES * 256;
    k_edgeE<<<(int)((nE + TB - 1) / TB), TB, 0, stream>>>(src, dst, ebuf, feat, attnsum);

    // per-node gate + head-average
    k_node<<<(NNODES + TB - 1) / TB, TB, 0, stream>>>(
        x, sumx, maxz, deg, W_gate, b_gate, attnsum, b_gat, gated);

    // merge GEMM: out = concat([x, gated]) @ W_merge + b_merge (K = 192)
    k_wmma_gemm<<<dim3(3125, 1), gemmBlk, 0, stream>>>(
        x, INFEAT, gated, OUTF, W_merge, b_merge, out, NNODES, OUTF);

    (void)in_sizes; (void)n_in; (void)out_size; (void)ws_size;
}